// SpectralConv3d_75127567941910
// MI455X (gfx1250) — compile-verified
//
#include <hip/hip_runtime.h>
#include <math.h>

// ---------------------------------------------------------------------------
// SpectralConv3d for MI455X (gfx1250, wave32).
// All DFT stages and the per-mode channel-mixing einsum are expressed as
// f32 GEMMs on V_WMMA_F32_16X16X4_F32.  Twiddle tables are stored in LDS in
// *fragment layout* (v2f pairs over even/odd k) so every B-operand is one
// aligned ds_load_b64 directly into the WMMA source VGPR pair.
// Sizes: B=4, Ci=Co=32, S=64, modes m=12 (kx,ky in {0..11, 52..63}, kz 0..11)
// ---------------------------------------------------------------------------

typedef __attribute__((ext_vector_type(2))) float v2f;
typedef __attribute__((ext_vector_type(8))) float v8f;

#define PI_F 3.14159265358979323846f

static __device__ __forceinline__ v8f wmma4(v2f a, v2f b, v8f c) {
  // D = A(16x4) * B(4x16) + C(16x16), fp32
  return __builtin_amdgcn_wmma_f32_16x16x4_f32(false, a, false, b,
                                               (short)0, c, false, false);
}

// mode index -> actual frequency for the 24-mode axes
static __device__ __forceinline__ int fmode(int n) { return (n < 12) ? n : (n + 40); }

// ---------------------------------------------------------------------------
// Stage 1: real -> complex truncated DFT along z (64 -> 12 modes).
// rows = (b,i,x,y) = 524288, K=64, N=12 (padded to 16).
// ---------------------------------------------------------------------------
__global__ void k_dftz(const float* __restrict__ x, float2* __restrict__ xz) {
  __shared__ v2f Tc[32 * 16];   // [k/2][n] -> {cos(2k..), cos(2k+1..)}
  __shared__ v2f Ts[32 * 16];
  for (int t = threadIdx.x; t < 32 * 16; t += blockDim.x) {
    int kp = t >> 4, n = t & 15;
    v2f ec = {}, es = {};
    if (n < 12) {
      float s0, c0, s1, c1;
      __sincosf(2.0f * PI_F * (float)(((2 * kp) * n) & 63) / 64.0f, &s0, &c0);
      __sincosf(2.0f * PI_F * (float)(((2 * kp + 1) * n) & 63) / 64.0f, &s1, &c1);
      ec.x = c0; ec.y = c1; es.x = s0; es.y = s1;
    }
    Tc[t] = ec; Ts[t] = es;
  }
  __syncthreads();
  int wave = threadIdx.x >> 5, lane = threadIdx.x & 31;
  int m = lane & 15, half = lane >> 4;
  int tile = blockIdx.x * 8 + wave;
  int row0 = tile * 16;
  const float* src = x + (size_t)(row0 + m) * 64;
  __builtin_prefetch(src + 64, 0, 1);
  v8f acc_re = {}; v8f acc_im = {};
#pragma unroll
  for (int ks = 0; ks < 16; ++ks) {
    int klo = ks * 4 + 2 * half;
    v2f a = *(const v2f*)(src + klo);
    v2f bc = Tc[(klo >> 1) * 16 + m];
    v2f bs = Ts[(klo >> 1) * 16 + m];
    acc_re = wmma4(a, bc, acc_re);
    acc_im = wmma4(a, bs, acc_im);
  }
  if (m < 12) {
#pragma unroll
    for (int r = 0; r < 8; ++r) {
      int mr = r + 8 * half;
      float2 o; o.x = acc_re[r]; o.y = -acc_im[r];   // e^{-i..}
      xz[(size_t)(row0 + mr) * 12 + m] = o;
    }
  }
}

// ---------------------------------------------------------------------------
// Generic complex -> complex truncated DFT along an axis of length 64,
// 24 modes {0..11, 52..63}.  Input [O][64][inner], output [O][24][inner].
// ---------------------------------------------------------------------------
__global__ void k_dftc(const float2* __restrict__ in, float2* __restrict__ out,
                       int inner) {
  __shared__ v2f Tc[32 * 32];
  __shared__ v2f Ts[32 * 32];
  for (int t = threadIdx.x; t < 32 * 32; t += blockDim.x) {
    int kp = t >> 5, n = t & 31;
    v2f ec = {}, es = {};
    if (n < 24) {
      int km = fmode(n);
      float s0, c0, s1, c1;
      __sincosf(2.0f * PI_F * (float)(((2 * kp) * km) & 63) / 64.0f, &s0, &c0);
      __sincosf(2.0f * PI_F * (float)(((2 * kp + 1) * km) & 63) / 64.0f, &s1, &c1);
      ec.x = c0; ec.y = c1; es.x = s0; es.y = s1;
    }
    Tc[t] = ec; Ts[t] = es;
  }
  __syncthreads();
  int wave = threadIdx.x >> 5, lane = threadIdx.x & 31;
  int m = lane & 15, half = lane >> 4;
  int tile = blockIdx.x * 4 + (wave >> 1);
  int nt = wave & 1;
  int row = tile * 16 + m;
  int o = row / inner, ii = row % inner;
  const float2* src = in + (size_t)o * 64 * inner + ii;
  int ncol = nt * 16 + m;
  v8f acc_re = {}; v8f acc_im = {};
#pragma unroll
  for (int ks = 0; ks < 16; ++ks) {
    int klo = ks * 4 + 2 * half;
    float2 c0 = src[(size_t)klo * inner];
    float2 c1 = src[(size_t)(klo + 1) * inner];
    v2f ar; ar.x = c0.x; ar.y = c1.x;
    v2f ai; ai.x = c0.y; ai.y = c1.y;
    v2f arn = -ar;
    v2f bc = Tc[(klo >> 1) * 32 + ncol];
    v2f bs = Ts[(klo >> 1) * 32 + ncol];
    acc_re = wmma4(ar, bc, acc_re);   // re =  ar*c + ai*s
    acc_re = wmma4(ai, bs, acc_re);
    acc_im = wmma4(ai, bc, acc_im);   // im =  ai*c - ar*s
    acc_im = wmma4(arn, bs, acc_im);
  }
  if (ncol < 24) {
#pragma unroll
    for (int r = 0; r < 8; ++r) {
      int mr = r + 8 * half;
      int rw = tile * 16 + mr;
      int oo = rw / inner, iii = rw % inner;
      float2 d; d.x = acc_re[r]; d.y = acc_im[r];
      out[((size_t)oo * 24 + ncol) * inner + iii] = d;
    }
  }
}

// ---------------------------------------------------------------------------
// x_ht = X + e^{i*2pi(kx+ky+kz)/64} * conj(X)   (mirror FFT folded in)
// over XH [b,i][24][24][12] complex, in place.  884736 elements.
// ---------------------------------------------------------------------------
__global__ void k_htcombine(float2* __restrict__ xh) {
  int tid = blockIdx.x * blockDim.x + threadIdx.x;
  int c  = tid % 12;
  int ky = (tid / 12) % 24;
  int kx = (tid / 288) % 24;
  float th = 2.0f * PI_F * (float)((fmode(kx) + fmode(ky) + c) & 63) / 64.0f;
  float s, cc; __sincosf(th, &s, &cc);
  float2 X = xh[tid];
  float2 H;
  H.x = X.x + cc * X.x + s * X.y;
  H.y = X.y + s * X.x - cc * X.y;
  xh[tid] = H;
}

// ---------------------------------------------------------------------------
// Weight prep: Y = dht(w) along last axis; Yplus/Yminus with flip-roll of w.
// Outputs WP/WM at [q][i][o][p], p=(a*12+b2)*12+k.  7,077,888 threads.
// ---------------------------------------------------------------------------
__global__ void k_wprep(const float* __restrict__ w, float2* __restrict__ wp,
                        float2* __restrict__ wm) {
  int tid = blockIdx.x * blockDim.x + threadIdx.x;
  int k  = tid % 12;
  int b2 = (tid / 12) % 12;
  int a  = (tid / 144) % 12;
  int o  = (tid / 1728) % 32;
  int i  = (tid / (1728 * 32)) % 32;
  int q  = tid / (1728 * 1024);
  const float* wq = w + ((((size_t)q * 32 + i) * 32 + o) * 144 + a * 12 + b2) * 24;
  float Y = 0.f;
  for (int c = 0; c < 12; ++c) {
    float s, cs; __sincosf(2.0f * PI_F * (float)((k * c) % 12) / 12.0f, &s, &cs);
    Y += wq[2 * c] * (cs + s) + wq[2 * c + 1] * (s - cs);  // Re - Im of FFT
  }
  int fi = (32 - i) & 31, fo = (32 - o) & 31, fa = (12 - a) % 12;
  const float* wf =
      w + ((((size_t)q * 32 + fi) * 32 + fo) * 144 + fa * 12 + b2) * 24 + 2 * k;
  float fr = wf[0], fim = wf[1];
  size_t idx = (((size_t)q * 32 + i) * 32 + o) * 1728 + (a * 12 + b2) * 12 + k;
  float2 P;  P.x = 0.5f * (Y + fr);  P.y = 0.5f * fim;
  float2 Mn; Mn.x = 0.5f * (Y - fr); Mn.y = -0.5f * fim;
  wp[idx] = P; wm[idx] = Mn;
}

// ---------------------------------------------------------------------------
// Mode mixing: per mode point p and block q, complex GEMM
//   Z[b,o] = sum_i x[b,i]*Yplus[i,o] + Xflip[b,i]*Yminus[i,o]
// M=4 (batch, rows 4..15 zero), K=32 (i), N=32 (o) -> 8 real WMMAs / K-step.
// grid.x = 1728 (p), 8 waves = q(4) x ntile(2).
// ---------------------------------------------------------------------------
__global__ void k_mix(const float2* __restrict__ xh, const float2* __restrict__ wp,
                      const float2* __restrict__ wm, float2* __restrict__ zm) {
  int p = blockIdx.x;
  int wave = threadIdx.x >> 5, lane = threadIdx.x & 31;
  int q = wave >> 1, nt = wave & 1;
  int m = lane & 15, half = lane >> 4;
  int c = p % 12, b2 = (p / 12) % 12, a = p / 144;
  int kxi = (q & 1) ? a + 12 : a;
  int kyi = (q & 2) ? b2 + 12 : b2;
  int fa = (12 - a) % 12;
  int fkxi = (q & 1) ? fa + 12 : fa;
  int fb = (4 - (m & 3)) & 3;
  int ocol = nt * 16 + m;   // 0..31, always valid
  bool mv = (m < 4);
  v8f acc_re = {}; v8f acc_im = {};
#pragma unroll
  for (int ks = 0; ks < 8; ++ks) {
    int i0 = ks * 4 + 2 * half;
    v2f ar = {}, ai = {}, fr = {}, fi2 = {};
    if (mv) {
#pragma unroll
      for (int j = 0; j < 2; ++j) {
        int i = i0 + j;
        float2 xv = xh[((((size_t)m * 32 + i) * 24 + kxi) * 24 + kyi) * 12 + c];
        int ff = (32 - i) & 31;
        float2 fv =
            xh[((((size_t)fb * 32 + ff) * 24 + fkxi) * 24 + kyi) * 12 + c];
        if (j == 0) { ar.x = xv.x; ai.x = xv.y; fr.x = fv.x; fi2.x = fv.y; }
        else        { ar.y = xv.x; ai.y = xv.y; fr.y = fv.x; fi2.y = fv.y; }
      }
    }
    v2f ain = -ai, fin = -fi2;
    size_t wb0 = (((size_t)q * 32 + i0) * 32 + ocol) * 1728 + p;
    size_t wb1 = (((size_t)q * 32 + i0 + 1) * 32 + ocol) * 1728 + p;
    float2 p0 = wp[wb0], p1 = wp[wb1];
    float2 m0 = wm[wb0], m1 = wm[wb1];
    v2f pr; pr.x = p0.x; pr.y = p1.x;
    v2f pi; pi.x = p0.y; pi.y = p1.y;
    v2f mr; mr.x = m0.x; mr.y = m1.x;
    v2f mi; mi.x = m0.y; mi.y = m1.y;
    acc_re = wmma4(ar,  pr, acc_re);
    acc_re = wmma4(ain, pi, acc_re);
    acc_im = wmma4(ar,  pi, acc_im);
    acc_im = wmma4(ai,  pr, acc_im);
    acc_re = wmma4(fr,  mr, acc_re);
    acc_re = wmma4(fin, mi, acc_re);
    acc_im = wmma4(fr,  mi, acc_im);
    acc_im = wmma4(fi2, mr, acc_im);
  }
  if (half == 0) {
#pragma unroll
    for (int r = 0; r < 4; ++r) {
      float2 z; z.x = acc_re[r]; z.y = acc_im[r];
      zm[(((size_t)q * 4 + r) * 32 + ocol) * 1728 + p] = z;
    }
  }
}

// ---------------------------------------------------------------------------
// dht(Z)/n along c (12-pt), scatter corners into V [b,o][24][24][12] real.
// ---------------------------------------------------------------------------
__global__ void k_zdht(const float2* __restrict__ zm, float* __restrict__ v) {
  int tid = blockIdx.x * blockDim.x + threadIdx.x;   // 884736
  int k  = tid % 12;
  int b2 = (tid / 12) % 12;
  int a  = (tid / 144) % 12;
  int o  = (tid / 1728) % 32;
  int b  = (tid / (1728 * 32)) % 4;
  int q  = tid / (1728 * 128);
  const float2* zp = zm + (((size_t)q * 4 + b) * 32 + o) * 1728 + (a * 12 + b2) * 12;
  float acc = 0.f;
  for (int c = 0; c < 12; ++c) {
    float s, cs; __sincosf(2.0f * PI_F * (float)((k * c) % 12) / 12.0f, &s, &cs);
    float2 z = zp[c];
    acc += z.x * (cs + s) + z.y * (s - cs);
  }
  acc *= (1.0f / 221184.0f);   // n = 4*32*12^3
  int kxi = (q & 1) ? a + 12 : a;
  int kyi = (q & 2) ? b2 + 12 : b2;
  v[((((size_t)b * 32 + o) * 24 + kxi) * 24 + kyi) * 12 + k] = acc;
}

// ---------------------------------------------------------------------------
// Inverse z (Hermitian cosine): 12 real modes -> 64 real, scale 1/64.
// rows = (b,o,kx,ky) = 73728, K=12, N=64 (4 n-tiles).
// ---------------------------------------------------------------------------
__global__ void k_invz(const float* __restrict__ v, float* __restrict__ oz) {
  __shared__ v2f Tc[6 * 64];    // [k/2][z]
  for (int t = threadIdx.x; t < 6 * 64; t += blockDim.x) {
    int kp = t >> 6, z = t & 63;
    int k0 = 2 * kp, k1 = 2 * kp + 1;
    float s, c0, c1;
    __sincosf(2.0f * PI_F * (float)((k0 * z) & 63) / 64.0f, &s, &c0);
    __sincosf(2.0f * PI_F * (float)((k1 * z) & 63) / 64.0f, &s, &c1);
    v2f e;
    e.x = ((k0 == 0) ? 1.0f : 2.0f) * c0 * (1.0f / 64.0f);
    e.y = 2.0f * c1 * (1.0f / 64.0f);
    Tc[t] = e;
  }
  __syncthreads();
  int wave = threadIdx.x >> 5, lane = threadIdx.x & 31;
  int m = lane & 15, half = lane >> 4;
  int tile = blockIdx.x * 2 + (wave >> 2);
  int nt = wave & 3;
  const float* src = v + (size_t)(tile * 16 + m) * 12;
  int zc = nt * 16 + m;
  v8f acc = {};
#pragma unroll
  for (int ks = 0; ks < 3; ++ks) {
    int klo = ks * 4 + 2 * half;
    v2f a = *(const v2f*)(src + klo);
    v2f bc = Tc[(klo >> 1) * 64 + zc];
    acc = wmma4(a, bc, acc);
  }
#pragma unroll
  for (int r = 0; r < 8; ++r) {
    int mr = r + 8 * half;
    oz[(size_t)(tile * 16 + mr) * 64 + zc] = acc[r];
  }
}

// ---------------------------------------------------------------------------
// Inverse y: real [o3][24][64] -> complex [o3][64][64], e^{+i..}/64.
// rows = (o3=b,o,kx; z) = 196608, K=24, N=64.
// ---------------------------------------------------------------------------
__global__ void k_invy(const float* __restrict__ iz, float2* __restrict__ iy) {
  __shared__ v2f Tc[12 * 64];   // [k/2][y]
  __shared__ v2f Ts[12 * 64];
  for (int t = threadIdx.x; t < 12 * 64; t += blockDim.x) {
    int kp = t >> 6, y = t & 63;
    float s0, c0, s1, c1;
    __sincosf(2.0f * PI_F * (float)((fmode(2 * kp) * y) & 63) / 64.0f, &s0, &c0);
    __sincosf(2.0f * PI_F * (float)((fmode(2 * kp + 1) * y) & 63) / 64.0f, &s1, &c1);
    v2f ec, es;
    ec.x = c0 * (1.0f / 64.0f); ec.y = c1 * (1.0f / 64.0f);
    es.x = s0 * (1.0f / 64.0f); es.y = s1 * (1.0f / 64.0f);
    Tc[t] = ec; Ts[t] = es;
  }
  __syncthreads();
  int wave = threadIdx.x >> 5, lane = threadIdx.x & 31;
  int m = lane & 15, half = lane >> 4;
  int tile = blockIdx.x * 2 + (wave >> 2);
  int nt = wave & 3;
  int row = tile * 16 + m;
  int o3 = row >> 6, z = row & 63;
  const float* src = iz + (size_t)o3 * 24 * 64 + z;
  int yc = nt * 16 + m;
  v8f acc_re = {}; v8f acc_im = {};
#pragma unroll
  for (int ks = 0; ks < 6; ++ks) {
    int klo = ks * 4 + 2 * half;
    v2f a; a.x = src[(size_t)klo * 64]; a.y = src[(size_t)(klo + 1) * 64];
    v2f bc = Tc[(klo >> 1) * 64 + yc];
    v2f bs = Ts[(klo >> 1) * 64 + yc];
    acc_re = wmma4(a, bc, acc_re);
    acc_im = wmma4(a, bs, acc_im);
  }
#pragma unroll
  for (int r = 0; r < 8; ++r) {
    int mr = r + 8 * half;
    int rw = tile * 16 + mr;
    int oo = rw >> 6, zz = rw & 63;
    float2 d; d.x = acc_re[r]; d.y = acc_im[r];
    iy[(((size_t)oo * 64) + yc) * 64 + zz] = d;
  }
}

// ---------------------------------------------------------------------------
// Inverse x: complex [bo][24][4096] -> real output [bo][64][4096], keep Re.
// rows = (bo; yz) = 524288, K=24, N=64.
// ---------------------------------------------------------------------------
__global__ void k_invx(const float2* __restrict__ iy, float* __restrict__ out) {
  __shared__ v2f Tc[12 * 64];
  __shared__ v2f Ts[12 * 64];
  for (int t = threadIdx.x; t < 12 * 64; t += blockDim.x) {
    int kp = t >> 6, xx = t & 63;
    float s0, c0, s1, c1;
    __sincosf(2.0f * PI_F * (float)((fmode(2 * kp) * xx) & 63) / 64.0f, &s0, &c0);
    __sincosf(2.0f * PI_F * (float)((fmode(2 * kp + 1) * xx) & 63) / 64.0f, &s1, &c1);
    v2f ec, es;
    ec.x = c0 * (1.0f / 64.0f); ec.y = c1 * (1.0f / 64.0f);
    es.x = s0 * (1.0f / 64.0f); es.y = s1 * (1.0f / 64.0f);
    Tc[t] = ec; Ts[t] = es;
  }
  __syncthreads();
  int wave = threadIdx.x >> 5, lane = threadIdx.x & 31;
  int m = lane & 15, half = lane >> 4;
  int tile = blockIdx.x * 2 + (wave >> 2);
  int nt = wave & 3;
  int row = tile * 16 + m;
  int bo = row >> 12, yz = row & 4095;
  const float2* src = iy + (size_t)bo * 24 * 4096 + yz;
  __builtin_prefetch(src, 0, 1);
  int xc = nt * 16 + m;
  v8f acc = {};
#pragma unroll
  for (int ks = 0; ks < 6; ++ks) {
    int klo = ks * 4 + 2 * half;
    float2 c0 = src[(size_t)klo * 4096];
    float2 c1 = src[(size_t)(klo + 1) * 4096];
    v2f vr;  vr.x = c0.x;  vr.y = c1.x;
    v2f vin; vin.x = -c0.y; vin.y = -c1.y;   // Re{(Vr+iVi)e^{i..}} = Vr c - Vi s
    v2f bc = Tc[(klo >> 1) * 64 + xc];
    v2f bs = Ts[(klo >> 1) * 64 + xc];
    acc = wmma4(vr, bc, acc);
    acc = wmma4(vin, bs, acc);
  }
#pragma unroll
  for (int r = 0; r < 8; ++r) {
    int mr = r + 8 * half;
    int rw = tile * 16 + mr;
    int bo2 = rw >> 12, yz2 = rw & 4095;
    out[(((size_t)bo2 * 64) + xc) * 4096 + yz2] = acc[r];
  }
}

// ---------------------------------------------------------------------------

extern "C" void kernel_launch(void* const* d_in, const int* in_sizes, int n_in,
                              void* d_out, int out_size, void* d_ws, size_t ws_size,
                              hipStream_t stream) {
  (void)in_sizes; (void)n_in; (void)out_size; (void)ws_size;
  const float* x = (const float*)d_in[0];
  const float* w = (const float*)d_in[1];
  float* out = (float*)d_out;
  float* ws  = (float*)d_ws;

  // workspace layout (float offsets)
  float2* XZ = (float2*)(ws + 0);          // [524288][12] cplx      (50.3 MB)
  float2* XY = (float2*)(ws + 12582912);   // [8192][24][12] cplx    (18.9 MB)
  float2* XH = (float2*)(ws + 17301504);   // [128][24][24][12] cplx ( 7.1 MB)
  float2* WP = (float2*)(ws + 19070976);   // [4][32][32][1728] cplx (56.6 MB)
  float2* WM = (float2*)(ws + 33226752);   // same                   (56.6 MB)
  float2* ZM = (float2*)(ws + 47382528);   // [4][4][32][1728] cplx  ( 7.1 MB)
  float*  V  = (float*)(ws + 49152000);    // [4][32][24][24][12]    ( 3.5 MB)
  float*  IZ = (float*)(ws + 50036736);    // [73728][64]            (18.9 MB)
  float2* IY = (float2*)(ws + 0);          // reuse dead fwd region  (100.7 MB)

  k_dftz<<<4096, 256, 0, stream>>>(x, XZ);              // z:  64 -> 12
  k_dftc<<<1536, 256, 0, stream>>>(XZ, XY, 12);         // y:  64 -> 24
  k_dftc<<<576,  256, 0, stream>>>(XY, XH, 288);        // x:  64 -> 24
  k_htcombine<<<884736 / 256, 256, 0, stream>>>(XH);    // + mirror via conj
  k_wprep<<<27648, 256, 0, stream>>>(w, WP, WM);        // Yplus / Yminus
  k_mix<<<1728, 256, 0, stream>>>(XH, WP, WM, ZM);      // channel mixing
  k_zdht<<<3456, 256, 0, stream>>>(ZM, V);              // dht(Z)/n -> corners
  k_invz<<<2304, 256, 0, stream>>>(V, IZ);              // z^-1: 12 -> 64
  k_invy<<<6144, 256, 0, stream>>>(IZ, IY);             // y^-1: 24 -> 64
  k_invx<<<16384, 256, 0, stream>>>(IY, out);           // x^-1: 24 -> 64, Re
}